// kspaceMap_5557687681593
// MI455X (gfx1250) — compile-verified
//
#include <hip/hip_runtime.h>
#include <math.h>

typedef __attribute__((ext_vector_type(2))) float v2f;
typedef __attribute__((ext_vector_type(8))) float v8f;

#define S 128

// ---------------------------------------------------------------------------
// Stage 0: twiddle tables  cosT[i,j] = cos(2*pi*i*j/S), sinT likewise.
// 128x128 each = 64KB; stays hot in L2 (192MB).
// ---------------------------------------------------------------------------
__global__ void build_tables_k(float* __restrict__ cosT, float* __restrict__ sinT) {
    int i = blockIdx.x * blockDim.x + threadIdx.x;   // 0..16383
    int r = i >> 7;
    int c = i & (S - 1);
    int t = (r * c) & (S - 1);                       // exploit period S for accuracy
    float ang = (float)t * (6.28318530717958647692f / (float)S);
    cosT[i] = cosf(ang);
    sinT[i] = sinf(ang);
}

// ---------------------------------------------------------------------------
// Stage 1: per-row real DFT as two fp32 WMMA GEMMs.
//   Fre[g,m] = sum_v x[g,v] * cosT[v,m]
//   Fsn[g,m] = sum_v x[g,v] * sinT[v,m]
// g in [0,2048) indexes (b,y); x row lives at input[(b*2)*S*S + y*S].
// One wave per 16x16 tile; blockDim=256 (8 waves) covers all 8 N-tiles,
// gridDim.x=128 covers the 128 M-tiles. 32 k-steps of V_WMMA_F32_16X16X4_F32
// per accumulator.
// ---------------------------------------------------------------------------
__global__ void __launch_bounds__(256) dft_rows_wmma_k(
    const float* __restrict__ x,        // (16,2,128,128), channel 0 used
    const float* __restrict__ cosT,
    const float* __restrict__ sinT,
    float* __restrict__ Fre,            // (2048,128)
    float* __restrict__ Fsn)            // (2048,128)
{
    const int lane = threadIdx.x & 31;
    const int wave = threadIdx.x >> 5;      // 0..7 -> N tile
    const int row0 = blockIdx.x * 16;       // M tile base (global row g)
    const int col0 = wave * 16;             // N tile base (m)
    const int lm   = lane & 15;             // M (for A) / N (for B,D) within tile
    const int hi   = lane >> 4;             // selects K pair {0,1} vs {2,3}

    // Row pointer for A fragment: skip channel-1 planes of the input.
    const int g = row0 + lm;                        // (b,y) combined
    const float* arow = x + (((g >> 7) * 2 * S) + (g & (S - 1))) * S;

    v8f accC = {};
    v8f accS = {};

    for (int k = 0; k < S; k += 4) {
        const int ka = k + 2 * hi;
        // A: 16x4 fp32 fragment. lane<16: K={k,k+1}; lane>=16: K={k+2,k+3}
        v2f a;
        a.x = arow[ka];
        a.y = arow[ka + 1];
        // B: 4x16 fp32 fragment (mirrored layout, N striped across lanes)
        v2f bc, bs;
        bc.x = cosT[ka * S + col0 + lm];
        bc.y = cosT[(ka + 1) * S + col0 + lm];
        bs.x = sinT[ka * S + col0 + lm];
        bs.y = sinT[(ka + 1) * S + col0 + lm];

        accC = __builtin_amdgcn_wmma_f32_16x16x4_f32(
            false, a, false, bc, (short)0, accC, false, false);
        accS = __builtin_amdgcn_wmma_f32_16x16x4_f32(
            false, a, false, bs, (short)0, accS, false, false);
    }

    // D layout: VGPR j -> M=j (lanes 0-15) / M=j+8 (lanes 16-31); N = lane&15
#pragma unroll
    for (int j = 0; j < 8; ++j) {
        const int gm = row0 + j + 8 * hi;
        const int n  = col0 + lm;
        Fre[gm * S + n] = accC[j];
        Fsn[gm * S + n] = accS[j];
    }
}

// ---------------------------------------------------------------------------
// Stage 2: out[b,m,y,x] = (Fre[b,y,m]*cosT[m,x] + Fsn[b,y,m]*sinT[m,x]) / S
// Pure streaming stage: 128MB of float4 stores -> HBM-bandwidth bound.
// One thread = 4 consecutive x. 8,388,608 threads.
// ---------------------------------------------------------------------------
__global__ void __launch_bounds__(256) modulate_k(
    const float* __restrict__ Fre, const float* __restrict__ Fsn,
    const float* __restrict__ cosT, const float* __restrict__ sinT,
    float* __restrict__ out)
{
    int t   = blockIdx.x * blockDim.x + threadIdx.x;
    int xi  = (t & 31) << 2;            // x base (32 float4 per row of 128)
    int row = t >> 5;                   // b*16384 + m*128 + y
    int y   = row & (S - 1);
    int m   = (row >> 7) & (S - 1);
    int b   = row >> 14;

    const int fidx = (((b << 7) + y) * S) + m;
    const float re = Fre[fidx];
    const float gn = Fsn[fidx];

    const float4 c4 = *reinterpret_cast<const float4*>(cosT + m * S + xi);
    const float4 s4 = *reinterpret_cast<const float4*>(sinT + m * S + xi);

    const float inv = 1.0f / (float)S;
    float4 o;
    o.x = (re * c4.x + gn * s4.x) * inv;
    o.y = (re * c4.y + gn * s4.y) * inv;
    o.z = (re * c4.z + gn * s4.z) * inv;
    o.w = (re * c4.w + gn * s4.w) * inv;

    *reinterpret_cast<float4*>(out + (((b * 129 + m) * S + y) * S) + xi) = o;
}

// ---------------------------------------------------------------------------
// Stage 3: copy input channel 1 into output channel 128.
// ---------------------------------------------------------------------------
__global__ void __launch_bounds__(256) copy_others_k(
    const float* __restrict__ in, float* __restrict__ out)
{
    int t   = blockIdx.x * blockDim.x + threadIdx.x;   // 65536 threads
    int idx = t << 2;                                  // flat float index
    int b   = idx >> 14;                               // / (128*128)
    int off = idx & 16383;
    float4 v = *reinterpret_cast<const float4*>(in + (b * 2 + 1) * 16384 + off);
    *reinterpret_cast<float4*>(out + (b * 129 + 128) * 16384 + off) = v;
}

// ---------------------------------------------------------------------------
extern "C" void kernel_launch(void* const* d_in, const int* in_sizes, int n_in,
                              void* d_out, int out_size, void* d_ws, size_t ws_size,
                              hipStream_t stream) {
    const float* in = (const float*)d_in[0];
    float* out = (float*)d_out;

    float* ws   = (float*)d_ws;
    float* cosT = ws;                    // 16384 floats
    float* sinT = ws + 16384;            // 16384 floats
    float* Fre  = ws + 32768;            // 2048*128 floats
    float* Fsn  = Fre + 2048 * 128;      // 2048*128 floats  (total ~2.2MB)

    build_tables_k<<<64, 256, 0, stream>>>(cosT, sinT);
    dft_rows_wmma_k<<<128, 256, 0, stream>>>(in, cosT, sinT, Fre, Fsn);
    modulate_k<<<32768, 256, 0, stream>>>(Fre, Fsn, cosT, sinT, out);
    copy_others_k<<<256, 256, 0, stream>>>(in, out);
}